// GenericHGTEncoder_52922587021517
// MI455X (gfx1250) — compile-verified
//
#include <hip/hip_runtime.h>
#include <hip/hip_bf16.h>

#define C 128
#define H 4
#define HD 32
#define N_DRUG 50000
#define N_DIS 20000
#define NEDGE 250000
#define EA_DIM 8

typedef __attribute__((ext_vector_type(2))) float v2f;
typedef __attribute__((ext_vector_type(8))) float v8f;

// ---------------------------------------------------------------------------
// sign-aware float atomic max via integer atomics (monotonic bit mapping)
// ---------------------------------------------------------------------------
__device__ __forceinline__ void atomicMaxFloat(float* addr, float v) {
  int iv = __float_as_int(v);
  if (iv >= 0) {
    atomicMax((int*)addr, iv);
  } else {
    atomicMin((unsigned int*)addr, (unsigned int)iv);
  }
}

// ---------------------------------------------------------------------------
// fill helper (segmax needs -inf, segsum/aggr need 0; ws is poisoned 0xAA)
// ---------------------------------------------------------------------------
__global__ void fill_f32(float* __restrict__ p, float v, int n) {
  int i = blockIdx.x * blockDim.x + threadIdx.x;
  if (i < n) p[i] = v;
}

// ---------------------------------------------------------------------------
// Y[nrows,128] = act(X[nrows,128] @ W[128,128] + b)
// mode 0: identity, mode 1: sigmoid
// One wave computes one 16x16 tile with V_WMMA_F32_16X16X4_F32 (K stepped by 4).
// 8 waves/block cover all 128 output columns of a 16-row stripe; the A-tile
// (16x128) is staged in LDS with coalesced loads.
//
// fp32 WMMA fragment layouts (CDNA5 ISA 7.12.2):
//   A 16x4 : lanes 0-15 -> M=lane, K={0,1} in v[0],v[1]; lanes 16-31 -> K={2,3}
//   B 4x16 : lanes 0-15 -> N=lane, K={0,1};             lanes 16-31 -> K={2,3}
//   C/D    : VGPR r, lanes 0-15 -> M=r, N=lane; lanes 16-31 -> M=r+8, N=lane-16
// ---------------------------------------------------------------------------
__global__ void linear128_wmma(const float* __restrict__ X,
                               const float* __restrict__ W,
                               const float* __restrict__ bias,
                               float* __restrict__ Y,
                               int nrows, int mode) {
  __shared__ float As[16][132];
  const int tid = threadIdx.x;
  const int r0  = blockIdx.x * 16;

#pragma unroll
  for (int i = 0; i < 8; ++i) {          // 16x128 cooperative load
    int flat = i * 256 + tid;
    int rr = flat >> 7, cc = flat & 127;
    As[rr][cc] = X[(size_t)(r0 + rr) * C + cc];
  }
  __syncthreads();

  const int lane = tid & 31;
  const int wid  = tid >> 5;
  const int c0   = wid * 16;             // this wave's 16 output columns
  const int m    = lane & 15;
  const int koff = (lane >> 4) << 1;     // 0 for lanes 0-15, 2 for lanes 16-31

  v8f acc = {};
#pragma unroll
  for (int kk = 0; kk < C; kk += 4) {
    v2f a, b;
    a[0] = As[m][kk + koff];
    a[1] = As[m][kk + koff + 1];
    b[0] = W[(size_t)(kk + koff) * C + c0 + m];
    b[1] = W[(size_t)(kk + koff + 1) * C + c0 + m];
    acc = __builtin_amdgcn_wmma_f32_16x16x4_f32(
        /*neg_a=*/false, a, /*neg_b=*/false, b,
        /*c_mod=*/(short)0, acc, /*reuse_a=*/false, /*reuse_b=*/false);
  }

  const float bv    = bias[c0 + m];
  const int   rbase = r0 + ((lane >> 4) << 3);
#pragma unroll
  for (int r = 0; r < 8; ++r) {
    float v = acc[r] + bv;
    if (mode == 1) v = 1.0f / (1.0f + expf(-v));
    Y[(size_t)(rbase + r) * C + c0 + m] = v;
  }
}

// ---------------------------------------------------------------------------
// Fused gate MLP: gate[e] = sigmoid( GELU(ea[e]@Wg1+bg1) @ Wg2 + bg2 )
// The 16x128 GELU(ea@Wg1) A-tile is computed straight into LDS (64 FMA/thread)
// instead of materializing a 128 MB Hg buffer -> saves 256 MB of HBM traffic
// and 128 MB of workspace. Then the same fp32 WMMA tile-GEMM against Wg2.
// ---------------------------------------------------------------------------
__global__ void gate_fused_wmma(const float* __restrict__ ea,
                                const float* __restrict__ Wg1,
                                const float* __restrict__ bg1,
                                const float* __restrict__ Wg2,
                                const float* __restrict__ bg2,
                                float* __restrict__ gate) {
  __shared__ float eaS[16][EA_DIM];
  __shared__ float As[16][132];
  const int tid = threadIdx.x;
  const int e0  = blockIdx.x * 16;

  if (tid < 16 * EA_DIM) {               // 16 edge-attr rows (512 B)
    int rr = tid >> 3, j = tid & 7;
    eaS[rr][j] = ea[(size_t)(e0 + rr) * EA_DIM + j];
  }
  __syncthreads();

#pragma unroll
  for (int i = 0; i < 8; ++i) {          // 2048 Hg elements, 8 per thread
    int flat = i * 256 + tid;
    int rr = flat >> 7, cc = flat & 127;
    float acc = bg1[cc];
#pragma unroll
    for (int j = 0; j < EA_DIM; ++j) acc += eaS[rr][j] * Wg1[j * C + cc];
    As[rr][cc] = 0.5f * acc * (1.0f + erff(acc * 0.7071067811865476f));  // gelu
  }
  __syncthreads();

  const int lane = tid & 31;
  const int wid  = tid >> 5;
  const int c0   = wid * 16;
  const int m    = lane & 15;
  const int koff = (lane >> 4) << 1;

  v8f acc = {};
#pragma unroll
  for (int kk = 0; kk < C; kk += 4) {
    v2f a, b;
    a[0] = As[m][kk + koff];
    a[1] = As[m][kk + koff + 1];
    b[0] = Wg2[(size_t)(kk + koff) * C + c0 + m];
    b[1] = Wg2[(size_t)(kk + koff + 1) * C + c0 + m];
    acc = __builtin_amdgcn_wmma_f32_16x16x4_f32(
        false, a, false, b, (short)0, acc, false, false);
  }

  const float bv    = bg2[c0 + m];
  const int   rbase = e0 + ((lane >> 4) << 3);
#pragma unroll
  for (int r = 0; r < 8; ++r) {
    float v = acc[r] + bv;
    v = 1.0f / (1.0f + expf(-v));        // sigmoid
    gate[(size_t)(rbase + r) * C + c0 + m] = v;
  }
}

// ---------------------------------------------------------------------------
// Pass A: per-edge attention logits + atomic segment-max.
// One wave per edge; lane l owns channels {l, l+32, l+64, l+96}; the k-th
// element belongs to head k with hd = lane, so each head reduces over the
// full wave (shuffle reductions).
// ---------------------------------------------------------------------------
__global__ void edge_logits_k(const float* __restrict__ srcT,
                              const float* __restrict__ dstT,
                              const float* __restrict__ gate,   // may be null
                              const int* __restrict__ si,
                              const int* __restrict__ di,
                              const float* __restrict__ attn,   // [H*HD] flat
                              float* __restrict__ logits,       // [E,H]
                              float* __restrict__ segmax,       // [n_dst,H]
                              int ne) {
  const int lane = threadIdx.x & 31;
  const int e = blockIdx.x * (blockDim.x >> 5) + (threadIdx.x >> 5);
  if (e >= ne) return;
  const int s = si[e], d = di[e];
  const float* sp = srcT + (size_t)s * C;
  const float* dp = dstT + (size_t)d * C;
  const float* gp = gate ? gate + (size_t)e * C : nullptr;

#pragma unroll
  for (int k = 0; k < H; ++k) {
    const int c = lane + 32 * k;
    float sv = sp[c], dv = dp[c];
    float mv = sv + dv;
    if (gp) mv *= gp[c];
    float v = sv * dv * 0.17677669529663687f   // 1/sqrt(HD)
            + mv * attn[32 * k + lane];
    for (int off = 16; off; off >>= 1) v += __shfl_xor(v, off, 32);
    if (lane == k) {
      logits[(size_t)e * H + k] = v;
      atomicMaxFloat(&segmax[(size_t)d * H + k], v);
    }
  }
}

// ---------------------------------------------------------------------------
// Pass B: ex = exp(logit - segmax); overwrite logits with ex; atomic segment-sum
// ---------------------------------------------------------------------------
__global__ void edge_expsum_k(float* __restrict__ logits,
                              const int* __restrict__ di,
                              const float* __restrict__ segmax,
                              float* __restrict__ segsum, int ne) {
  int idx = blockIdx.x * blockDim.x + threadIdx.x;
  if (idx >= ne * H) return;
  int e = idx >> 2, h = idx & 3;
  int d = di[e];
  float ex = expf(logits[idx] - segmax[(size_t)d * H + h]);
  logits[idx] = ex;
  atomicAdd(&segsum[(size_t)d * H + h], ex);
}

// ---------------------------------------------------------------------------
// Pass C: aggr[d] += msg * w   (recompute msg from L2-resident node tables)
// ---------------------------------------------------------------------------
__global__ void edge_scatter_k(const float* __restrict__ srcT,
                               const float* __restrict__ dstT,
                               const float* __restrict__ gate,  // may be null
                               const int* __restrict__ si,
                               const int* __restrict__ di,
                               const float* __restrict__ exw,   // ex in logits buf
                               const float* __restrict__ segsum,
                               float* __restrict__ aggr, int ne) {
  const int lane = threadIdx.x & 31;
  const int e = blockIdx.x * (blockDim.x >> 5) + (threadIdx.x >> 5);
  if (e >= ne) return;
  const int s = si[e], d = di[e];
  const float* sp = srcT + (size_t)s * C;
  const float* dp = dstT + (size_t)d * C;
  const float* gp = gate ? gate + (size_t)e * C : nullptr;

#pragma unroll
  for (int k = 0; k < H; ++k) {
    const int c = lane + 32 * k;
    float mv = sp[c] + dp[c];
    if (gp) mv *= gp[c];
    float w = exw[(size_t)e * H + k] / (segsum[(size_t)d * H + k] + 1e-16f);
    atomicAdd(&aggr[(size_t)d * C + c], mv * w);
  }
}

// ---------------------------------------------------------------------------
extern "C" void kernel_launch(void* const* d_in, const int* in_sizes, int n_in,
                              void* d_out, int out_size, void* d_ws, size_t ws_size,
                              hipStream_t stream) {
  const float* x_drug  = (const float*)d_in[0];
  const float* x_dis   = (const float*)d_in[1];
  const float* ea      = (const float*)d_in[2];
  const int*   si_t    = (const int*)d_in[3];
  const int*   di_t    = (const int*)d_in[4];
  const int*   si_r    = (const int*)d_in[5];
  const int*   di_r    = (const int*)d_in[6];
  const float* Ws_t    = (const float*)d_in[7];
  const float* bs_t    = (const float*)d_in[8];
  const float* Wd_t    = (const float*)d_in[9];
  const float* bd_t    = (const float*)d_in[10];
  const float* attn_t  = (const float*)d_in[11];
  const float* Wg1     = (const float*)d_in[12];
  const float* bg1     = (const float*)d_in[13];
  const float* Wg2     = (const float*)d_in[14];
  const float* bg2     = (const float*)d_in[15];
  const float* Ws_r    = (const float*)d_in[16];
  const float* bs_r    = (const float*)d_in[17];
  const float* Wd_r    = (const float*)d_in[18];
  const float* bd_r    = (const float*)d_in[19];
  const float* attn_r  = (const float*)d_in[20];
  const float* Wo_drug = (const float*)d_in[21];
  const float* bo_drug = (const float*)d_in[22];
  const float* Wo_dis  = (const float*)d_in[23];
  const float* bo_dis  = (const float*)d_in[24];
  float* out = (float*)d_out;

  // workspace layout (floats); peak ~241 MB
  float* ws = (float*)d_ws;
  size_t off = 0;
  float* drugS   = ws + off; off += (size_t)N_DRUG * C;  // x_drug @ Ws_t + bs_t
  float* disD    = ws + off; off += (size_t)N_DIS  * C;  // x_dis  @ Wd_t + bd_t
  float* disS    = ws + off; off += (size_t)N_DIS  * C;  // x_dis  @ Ws_r + bs_r
  float* drugD   = ws + off; off += (size_t)N_DRUG * C;  // x_drug @ Wd_r + bd_r
  float* gate    = ws + off; off += (size_t)NEDGE  * C;  // fused gate MLP output
  float* lgts    = ws + off; off += (size_t)NEDGE  * H;  // logits -> ex (in place)
  float* segmax  = ws + off; off += (size_t)N_DRUG * H;  // sized for larger side
  float* segsum  = ws + off; off += (size_t)N_DRUG * H;
  float* aggrDis = ws + off; off += (size_t)N_DIS  * C;
  float* aggrDrg = ws + off; off += (size_t)N_DRUG * C;

  const dim3 blk(256);
  const float NEG_INF = -__builtin_huge_valf();

  // hoisted node-level linear transforms (WMMA f32)
  linear128_wmma<<<N_DRUG / 16, blk, 0, stream>>>(x_drug, Ws_t, bs_t, drugS, N_DRUG, 0);
  linear128_wmma<<<N_DIS  / 16, blk, 0, stream>>>(x_dis,  Wd_t, bd_t, disD,  N_DIS,  0);
  linear128_wmma<<<N_DIS  / 16, blk, 0, stream>>>(x_dis,  Ws_r, bs_r, disS,  N_DIS,  0);
  linear128_wmma<<<N_DRUG / 16, blk, 0, stream>>>(x_drug, Wd_r, bd_r, drugD, N_DRUG, 0);

  // fused edge-gate MLP (GELU stage computed into LDS, WMMA GEMM vs Wg2)
  gate_fused_wmma<<<NEDGE / 16, blk, 0, stream>>>(ea, Wg1, bg1, Wg2, bg2, gate);

  // direction 1: drug --treats--> disease (gated)
  fill_f32<<<(N_DIS * H + 255) / 256, blk, 0, stream>>>(segmax, NEG_INF, N_DIS * H);
  fill_f32<<<(N_DIS * H + 255) / 256, blk, 0, stream>>>(segsum, 0.0f, N_DIS * H);
  fill_f32<<<(N_DIS * C + 255) / 256, blk, 0, stream>>>(aggrDis, 0.0f, N_DIS * C);
  edge_logits_k<<<(NEDGE + 7) / 8, blk, 0, stream>>>(drugS, disD, gate, si_t, di_t,
                                                     attn_t, lgts, segmax, NEDGE);
  edge_expsum_k<<<(NEDGE * H + 255) / 256, blk, 0, stream>>>(lgts, di_t, segmax, segsum, NEDGE);
  edge_scatter_k<<<(NEDGE + 7) / 8, blk, 0, stream>>>(drugS, disD, gate, si_t, di_t,
                                                      lgts, segsum, aggrDis, NEDGE);

  // direction 2: disease --rev--> drug (ungated)
  fill_f32<<<(N_DRUG * H + 255) / 256, blk, 0, stream>>>(segmax, NEG_INF, N_DRUG * H);
  fill_f32<<<(N_DRUG * H + 255) / 256, blk, 0, stream>>>(segsum, 0.0f, N_DRUG * H);
  fill_f32<<<(N_DRUG * C + 255) / 256, blk, 0, stream>>>(aggrDrg, 0.0f, N_DRUG * C);
  edge_logits_k<<<(NEDGE + 7) / 8, blk, 0, stream>>>(disS, drugD, nullptr, si_r, di_r,
                                                     attn_r, lgts, segmax, NEDGE);
  edge_expsum_k<<<(NEDGE * H + 255) / 256, blk, 0, stream>>>(lgts, di_r, segmax, segsum, NEDGE);
  edge_scatter_k<<<(NEDGE + 7) / 8, blk, 0, stream>>>(disS, drugD, nullptr, si_r, di_r,
                                                      lgts, segsum, aggrDrg, NEDGE);

  // output linears straight into d_out: [out_drug | out_dis]
  linear128_wmma<<<N_DRUG / 16, blk, 0, stream>>>(aggrDrg, Wo_drug, bo_drug, out, N_DRUG, 0);
  linear128_wmma<<<N_DIS  / 16, blk, 0, stream>>>(aggrDis, Wo_dis, bo_dis,
                                                  out + (size_t)N_DRUG * C, N_DIS, 0);
}